// MultiSimilarityLoss_12833362280682
// MI455X (gfx1250) — compile-verified
//
#include <hip/hip_runtime.h>
#include <hip/hip_bf16.h>
#include <math.h>

typedef __attribute__((ext_vector_type(2))) float v2f;
typedef __attribute__((ext_vector_type(8))) float v8f;

#define MS_ALPHA 2.0f
#define MS_BETA  50.0f
#define MS_LAM   0.5f
#define MS_EPS   1e-8f
#define MS_N     4096
#define MS_D     256
#define APAD     4          // LDS row pad: bank = (4*row + k) % 64 -> conflict free

__global__ void ms_zero_out(float* out) { out[0] = 0.0f; }

// One wave (32 lanes) normalizes one 256-float row. 8 rows per 256-thread block.
__global__ __launch_bounds__(256) void ms_normalize(const float* __restrict__ e,
                                                    float* __restrict__ z) {
    const int wave = threadIdx.x >> 5;
    const int lane = threadIdx.x & 31;
    const int row  = blockIdx.x * 8 + wave;
    const float4* src = (const float4*)(e + (size_t)row * MS_D);
    float4 x0 = src[lane];
    float4 x1 = src[lane + 32];
    float s = x0.x*x0.x + x0.y*x0.y + x0.z*x0.z + x0.w*x0.w
            + x1.x*x1.x + x1.y*x1.y + x1.z*x1.z + x1.w*x1.w;
    #pragma unroll
    for (int off = 16; off >= 1; off >>= 1) s += __shfl_xor(s, off, 32);
    const float inv = 1.0f / fmaxf(sqrtf(s), MS_EPS);
    float4* dst = (float4*)(z + (size_t)row * MS_D);
    x0.x *= inv; x0.y *= inv; x0.z *= inv; x0.w *= inv;
    x1.x *= inv; x1.y *= inv; x1.z *= inv; x1.w *= inv;
    dst[lane]      = x0;
    dst[lane + 32] = x1;
}

// One workgroup = 16 rows of sim. 8 waves split the 256 column tiles,
// each wave processing TWO tiles per iteration (shared A-fragments, two
// independent WMMA accumulator chains, 2x loads in flight).
__global__ __launch_bounds__(256) void ms_loss(const float* __restrict__ z,
                                               const int* __restrict__ labels,
                                               float* __restrict__ out) {
    __shared__ float As[16][MS_D + APAD];
    __shared__ float redP[8][16];
    __shared__ float redN[8][16];
    __shared__ int   labI[16];

    const int t  = threadIdx.x;
    const int i0 = blockIdx.x * 16;

    // Stage A tile (16 x 256 fp32) into padded LDS, coalesced float4 loads.
    #pragma unroll
    for (int i = 0; i < 4; ++i) {
        int idx  = t + i * 256;        // float4 index in [0,1024)
        int row  = idx >> 6;           // 64 float4 per row
        int c4   = idx & 63;
        float4 v = ((const float4*)(z + (size_t)(i0 + row) * MS_D))[c4];
        As[row][c4 * 4 + 0] = v.x;
        As[row][c4 * 4 + 1] = v.y;
        As[row][c4 * 4 + 2] = v.z;
        As[row][c4 * 4 + 3] = v.w;
    }
    if (t < 16) labI[t] = labels[i0 + t];
    __syncthreads();

    const int w    = t >> 5;       // wave id 0..7
    const int lane = t & 31;
    const int m    = lane & 15;    // M (for A) / N (for B/C) position
    const int h    = lane >> 4;    // half-wave selects K sub-pair / C row group

    float posAcc[8];
    float negAcc[8];
    #pragma unroll
    for (int r = 0; r < 8; ++r) { posAcc[r] = 0.0f; negAcc[r] = 0.0f; }

    const float* Ap = &As[m][2 * h];

    // Wave w handles tile pairs (jt, jt+8) for jt = w, w+16, ... (16 uniform iters).
    for (int jt = w; jt < 256; jt += 16) {
        const int j0a = jt * 16;
        const int j0b = (jt + 8) * 16;
        v8f c0 = {0.f, 0.f, 0.f, 0.f, 0.f, 0.f, 0.f, 0.f};
        v8f c1 = {0.f, 0.f, 0.f, 0.f, 0.f, 0.f, 0.f, 0.f};
        const float* BpA = z + (size_t)(j0a + m) * MS_D + 2 * h;
        const float* BpB = z + (size_t)(j0b + m) * MS_D + 2 * h;

        // Prefetch next iteration's B rows into cache (global_prefetch_b8).
        if (jt + 16 < 256) {
            __builtin_prefetch(BpA + 256 * MS_D, 0, 1);
            __builtin_prefetch(BpB + 256 * MS_D, 0, 1);
        }

        // 8 chunks of 8 K-steps: stage loads in clauses, then a 16-WMMA burst.
        for (int ch = 0; ch < 8; ++ch) {
            v2f a[8], b0[8], b1[8];
            #pragma unroll
            for (int u = 0; u < 8; ++u) {
                b0[u] = *(const v2f*)(BpA + 32 * ch + 4 * u);   // z[j0a+m][...]
                b1[u] = *(const v2f*)(BpB + 32 * ch + 4 * u);   // z[j0b+m][...]
            }
            #pragma unroll
            for (int u = 0; u < 8; ++u) {
                a[u] = *(const v2f*)(Ap + 32 * ch + 4 * u);     // As[m][...] (LDS)
            }
            #pragma unroll
            for (int u = 0; u < 8; ++u) {
                c0 = __builtin_amdgcn_wmma_f32_16x16x4_f32(
                         false, a[u], false, b0[u], (short)0, c0, false, false);
                c1 = __builtin_amdgcn_wmma_f32_16x16x4_f32(
                         false, a[u], false, b1[u], (short)0, c1, false, false);
            }
        }

        // c0[r] = sim[i0 + r + 8h][j0a + m], c1[r] = sim[i0 + r + 8h][j0b + m]
        const int lja = labels[j0a + m];
        const int ljb = labels[j0b + m];
        #pragma unroll
        for (int r = 0; r < 8; ++r) {
            const int row = r + 8 * h;
            {
                const bool same = (labI[row] == lja);
                const bool diag = (i0 + row) == (j0a + m);
                const float s   = c0[r];
                if (same && !diag) posAcc[r] += __expf(-MS_ALPHA * (s - MS_LAM));
                if (!same)         negAcc[r] += __expf( MS_BETA  * (s - MS_LAM));
            }
            {
                const bool same = (labI[row] == ljb);
                const bool diag = (i0 + row) == (j0b + m);
                const float s   = c1[r];
                if (same && !diag) posAcc[r] += __expf(-MS_ALPHA * (s - MS_LAM));
                if (!same)         negAcc[r] += __expf( MS_BETA  * (s - MS_LAM));
            }
        }
    }

    // Reduce across the 16 lanes of each half-wave (N dimension).
    #pragma unroll
    for (int r = 0; r < 8; ++r) {
        #pragma unroll
        for (int off = 8; off >= 1; off >>= 1) {
            posAcc[r] += __shfl_xor(posAcc[r], off, 32);
            negAcc[r] += __shfl_xor(negAcc[r], off, 32);
        }
    }
    if (m == 0) {
        #pragma unroll
        for (int r = 0; r < 8; ++r) {
            redP[w][r + 8 * h] = posAcc[r];
            redN[w][r + 8 * h] = negAcc[r];
        }
    }
    __syncthreads();

    // Cross-wave combine + per-row log1p + workgroup scalar reduce.
    if (t < 32) {
        float loss = 0.0f;
        if (lane < 16) {
            float ps = 0.0f, ns = 0.0f;
            #pragma unroll
            for (int ww = 0; ww < 8; ++ww) { ps += redP[ww][lane]; ns += redN[ww][lane]; }
            loss = log1pf(ps) + log1pf(ns);   // log1p(0)=0 covers has_pos/has_neg
        }
        #pragma unroll
        for (int off = 8; off >= 1; off >>= 1) loss += __shfl_xor(loss, off, 32);
        if (t == 0) atomicAdd(out, loss * (1.0f / (float)MS_N));
    }
}

extern "C" void kernel_launch(void* const* d_in, const int* in_sizes, int n_in,
                              void* d_out, int out_size, void* d_ws, size_t ws_size,
                              hipStream_t stream) {
    (void)in_sizes; (void)n_in; (void)out_size; (void)ws_size;
    const float* emb    = (const float*)d_in[0];
    const int*   labels = (const int*)d_in[1];
    float*       out    = (float*)d_out;
    float*       z      = (float*)d_ws;   // 4096*256*4 = 4 MB normalized embeddings

    ms_zero_out<<<1, 1, 0, stream>>>(out);
    ms_normalize<<<MS_N / 8, 256, 0, stream>>>(emb, z);
    ms_loss<<<MS_N / 16, 256, 0, stream>>>(z, labels, out);
}